// SetCriterion_2_28140625724039
// MI455X (gfx1250) — compile-verified
//
#include <hip/hip_runtime.h>
#include <math.h>

// Problem constants (from reference)
#define G      96
#define BATCH  64
#define NOBJ   32
#define KPT    63
#define MOBJ   2048            // BATCH*NOBJ
#define NCONF  589824          // BATCH*G*G
#define CONF_STRIDE 147456     // 576 blocks * 256 threads ; NCONF = 4*CONF_STRIDE exactly

// Workspace layout (float offsets)
#define OFF_PL    0
#define OFF_PT    2048
#define OFF_PR    4096
#define OFF_PB    6144
#define OFF_PA    8192
#define OFF_TL    10240
#define OFF_TT    12288
#define OFF_TR    14336
#define OFF_TB    16384
#define OFF_TA    18432
#define OFF_BOXP  20480        // 2048 per-object partials
#define OFF_KEYP  22528
#define OFF_CCP   24576
#define OFF_CONFP 26624        // 576 block partials
#define OFF_IOUP  27200        // 64 block partials

typedef float v2f __attribute__((ext_vector_type(2)));
typedef float v8f __attribute__((ext_vector_type(8)));

// ---- wave32 sum-reduction via V_WMMA_F32_16X16X4_F32 with all-ones B ----
// D[m][n] = sum_k A[m][k]  (B == ones, C == 0). Each lane's value occupies
// exactly one A slot (second A VGPR is zero), so summing one representative
// column of every row recovers the full 32-lane sum:
//   lanes 0-15 hold rows 0-7 in d[0..7], lanes 16-31 hold rows 8-15.
// Cross-half combine via v_readlane (wave-uniform, no LDS round-trip).
__device__ __forceinline__ float wave_reduce_wmma(float s) {
    v2f a; a.x = s;    a.y = 0.0f;
    v2f o; o.x = 1.0f; o.y = 1.0f;   // all-ones B: layout-independent
    v8f c = {};
    v8f d = __builtin_amdgcn_wmma_f32_16x16x4_f32(
        /*neg_a=*/false, a, /*neg_b=*/false, o,
        /*c_mod=*/(short)0, c, /*reuse_a=*/false, /*reuse_b=*/false);
    float r = ((d[0] + d[1]) + (d[2] + d[3])) + ((d[4] + d[5]) + (d[6] + d[7]));
    const float lo = __int_as_float(__builtin_amdgcn_readlane(__float_as_int(r), 0));
    const float hi = __int_as_float(__builtin_amdgcn_readlane(__float_as_int(r), 16));
    return lo + hi;
}

// Deterministic block reduction; result valid on thread 0 only.
template <int NT>
__device__ __forceinline__ float block_reduce(float v, float* lds) {
    float w = wave_reduce_wmma(v);
    const int lane = threadIdx.x & 31;
    const int wid  = threadIdx.x >> 5;
    if (lane == 0) lds[wid] = w;
    __syncthreads();
    float r = 0.0f;
    if (threadIdx.x == 0) {
#pragma unroll
        for (int i = 0; i < NT / 32; ++i) r += lds[i];
    }
    __syncthreads();   // make lds reusable by the next reduction
    return r;
}

__device__ __forceinline__ float softplusf(float x) {
    return fmaxf(x, 0.0f) + log1pf(__expf(-fabsf(x)));
}

// ---- Kernel 1: per-object gather + L1 partials + conf correction + box staging ----
// One 64-thread block (2 waves) per object.
__global__ void gather_kernel(const float* __restrict__ pb, const float* __restrict__ pc,
                              const float* __restrict__ pk, const float* __restrict__ tb,
                              const float* __restrict__ tk, float* __restrict__ ws) {
    __shared__ float lds[8];
    const int obj = blockIdx.x;          // b*NOBJ + n
    const int b   = obj >> 5;
    const int t   = threadIdx.x;

    const float* tbv = tb + (size_t)obj * 6;
    const float t_x = tbv[0] * (float)G;
    const float t_y = tbv[1] * (float)G;
    const float t_z = tbv[2];
    const float t_w = tbv[3] * (float)G;
    const float t_h = tbv[4] * (float)G;
    const float t_d = tbv[5];
    const int gi = (int)t_x;             // coords positive: trunc == floor
    const int gj = (int)t_y;
    const int cell = (b * G + gj) * G + gi;

    // keypoint L1 (63 values across 64 lanes)
    const float* kp  = pk + (size_t)cell * KPT;
    const float* tkp = tk + (size_t)obj * KPT;
    float keyv = (t < KPT) ? fabsf(kp[t] - tkp[t]) : 0.0f;

    float boxv = 0.0f, ccv = 0.0f;
    if (t == 0) {
        const float* p = pb + (size_t)cell * 6;
        const float x_o = p[0], y_o = p[1], z_o = p[2];
        const float w_o = p[3], h_o = p[4], d_o = p[5];
        const float cl  = pc[cell];

        const float tx = t_x - floorf(t_x);
        const float ty = t_y - floorf(t_y);
        const float tw = logf(t_w * (1.0f / 3.0f) + 1e-16f);
        const float th = logf(t_h * (1.0f / 3.0f) + 1e-16f);
        const float td = logf(t_d + 1e-16f);

        boxv = fabsf(x_o - tx) + fabsf(y_o - ty) + fabsf(w_o - tw)
             + fabsf(h_o - th) + fabsf(z_o - t_z) + fabsf(d_o - td);

        // BCE correction at positive cell: EMPTY_WEIGHT*sp(-c) - sp(c)
        ccv = 2.0f * softplusf(-cl) - softplusf(cl);

        // Stage grid-scale xyxy boxes + areas for the pairwise IoU pass
        const float px = x_o + (float)gi, py = y_o + (float)gj;
        const float pw = __expf(w_o) * 3.0f, ph = __expf(h_o) * 3.0f;
        const float pl = px - 0.5f * pw, pr = px + 0.5f * pw;
        const float pt = py - 0.5f * ph, pbm = py + 0.5f * ph;
        ws[OFF_PL + obj] = pl;  ws[OFF_PT + obj] = pt;
        ws[OFF_PR + obj] = pr;  ws[OFF_PB + obj] = pbm;
        ws[OFF_PA + obj] = (pr - pl) * (pbm - pt);

        const float tl = t_x - 0.5f * t_w, tr = t_x + 0.5f * t_w;
        const float tt = t_y - 0.5f * t_h, tbm = t_y + 0.5f * t_h;
        ws[OFF_TL + obj] = tl;  ws[OFF_TT + obj] = tt;
        ws[OFF_TR + obj] = tr;  ws[OFF_TB + obj] = tbm;
        ws[OFF_TA + obj] = (tr - tl) * (tbm - tt);
    }

    const float sb = block_reduce<64>(boxv, lds);
    const float sk = block_reduce<64>(keyv, lds);
    const float sc = block_reduce<64>(ccv,  lds);
    if (t == 0) {
        ws[OFF_BOXP + obj] = sb;
        ws[OFF_KEYP + obj] = sk;
        ws[OFF_CCP  + obj] = sc;
    }
}

// ---- Kernel 2: base BCE sum (softplus over whole grid) ----
__global__ void conf_kernel(const float* __restrict__ pc, float* __restrict__ out_partial) {
    __shared__ float lds[8];
    const int idx = blockIdx.x * 256 + threadIdx.x;
    float s = 0.0f;
#pragma unroll
    for (int k = 0; k < 4; ++k) s += softplusf(pc[idx + k * CONF_STRIDE]);
    const float r = block_reduce<256>(s, lds);
    if (threadIdx.x == 0) out_partial[blockIdx.x] = r;
}

// ---- Kernel 3: pairwise IoU, 256x256 tiles; target tile broadcast from LDS ----
__global__ void iou_kernel(const float* __restrict__ ws, float* __restrict__ out_partial) {
    __shared__ float st[5][256];
    __shared__ float lds[8];
    const int t  = threadIdx.x;
    const int bi = blockIdx.x >> 3;
    const int bj = blockIdx.x & 7;
    const int j0 = bj * 256;

    st[0][t] = ws[OFF_TL + j0 + t];
    st[1][t] = ws[OFF_TT + j0 + t];
    st[2][t] = ws[OFF_TR + j0 + t];
    st[3][t] = ws[OFF_TB + j0 + t];
    st[4][t] = ws[OFF_TA + j0 + t];
    __syncthreads();

    const int i = bi * 256 + t;
    const float l  = ws[OFF_PL + i];
    const float tt = ws[OFF_PT + i];
    const float r  = ws[OFF_PR + i];
    const float bb = ws[OFF_PB + i];
    const float a1 = ws[OFF_PA + i];

    float acc = 0.0f;
    for (int j = 0; j < 256; ++j) {               // LDS broadcast reads, no conflicts
        const float iw = fmaxf(fminf(r,  st[2][j]) - fmaxf(l,  st[0][j]), 0.0f);
        const float ih = fmaxf(fminf(bb, st[3][j]) - fmaxf(tt, st[1][j]), 0.0f);
        const float inter = iw * ih;
        const float uni   = a1 + st[4][j] - inter;
        acc += 1.0f - inter / uni;
    }
    const float s = block_reduce<256>(acc, lds);
    if (t == 0) out_partial[blockIdx.x] = s;
}

// ---- Kernel 4: deterministic final combine ----
__global__ void final_kernel(const float* __restrict__ ws, float* __restrict__ out) {
    __shared__ float lds[8];
    const int t = threadIdx.x;
    float v;

    v = 0.0f; for (int i = t; i < MOBJ; i += 256) v += ws[OFF_BOXP + i];
    const float sbox = block_reduce<256>(v, lds);
    v = 0.0f; for (int i = t; i < MOBJ; i += 256) v += ws[OFF_KEYP + i];
    const float skey = block_reduce<256>(v, lds);
    v = 0.0f; for (int i = t; i < MOBJ; i += 256) v += ws[OFF_CCP + i];
    const float scc  = block_reduce<256>(v, lds);
    v = 0.0f; for (int i = t; i < 576; i += 256) v += ws[OFF_CONFP + i];
    const float scnf = block_reduce<256>(v, lds);
    v = (t < 64) ? ws[OFF_IOUP + t] : 0.0f;
    const float siou = block_reduce<256>(v, lds);

    if (t == 0) {
        out[0] = sbox * (1.0f / (float)MOBJ);
        out[1] = skey * (1.0f / (float)MOBJ);
        out[2] = (scnf + scc) * (1.0f / (float)NCONF);
        out[3] = siou * (1.0f / (float)MOBJ);
    }
}

extern "C" void kernel_launch(void* const* d_in, const int* in_sizes, int n_in,
                              void* d_out, int out_size, void* d_ws, size_t ws_size,
                              hipStream_t stream) {
    (void)in_sizes; (void)n_in; (void)out_size; (void)ws_size;
    const float* pb = (const float*)d_in[0];   // pred_boxes      [64,96,96,6]
    const float* pc = (const float*)d_in[1];   // pred_confidence [64,96,96,1]
    const float* pk = (const float*)d_in[2];   // pred_keypoints  [64,96,96,63]
    const float* tb = (const float*)d_in[3];   // target_boxes    [64,32,6]
    const float* tk = (const float*)d_in[4];   // target_keypoints[64,32,63]
    float* ws  = (float*)d_ws;
    float* out = (float*)d_out;

    gather_kernel<<<MOBJ, 64, 0, stream>>>(pb, pc, pk, tb, tk, ws);
    conf_kernel<<<576, 256, 0, stream>>>(pc, ws + OFF_CONFP);
    iou_kernel<<<64, 256, 0, stream>>>(ws, ws + OFF_IOUP);
    final_kernel<<<1, 256, 0, stream>>>(ws, out);
}